// FilterDetection_90984587199189
// MI455X (gfx1250) — compile-verified
//
#include <hip/hip_runtime.h>

// Problem dims (from reference setup_inputs)
#define HDIM   1024
#define WDIM   1024
#define NBATCH 32

// Tiling
#define TW  128            // output tile width
#define TH  32             // output tile height
#define NT  256            // threads per block (8 wave32 waves)
#define TBH (TH + 7)       // 39 staged rows  (+/-3 halo)
#define CW  35             // 16B chunks per staged row (140 cols, gx0-4 .. gx0+135)
#define TCW (CW * 4)       // 140 staged cols
#define HC  (TW + 4)       // 132 cols after horizontal min
#define EH  (TH + 4)       // 36 rows after vertical min
#define SA  144            // LDS row stride (floats), 16B-aligned rows

__device__ __forceinline__ float thr(float v) { return v >= 0.5f ? v : 0.0f; }

__global__ __launch_bounds__(NT) void score_thresh_kernel(const float* __restrict__ s,
                                                          float* __restrict__ o, int n) {
  int i = blockIdx.x * NT + threadIdx.x;
  if (i < n) o[i] = thr(s[i]);
}

// Morphological opening (erode 4x4 offsets [-2,+1], then dilate 4x4 offsets
// [-1,+2]) of the thresholded mask; borders behave as cv2 (never win).
__global__ __launch_bounds__(NT) void open4_kernel(const float* __restrict__ mask,
                                                   float* __restrict__ out) {
  __shared__ float A[TBH * SA];   // raw tile  -> later eroded
  __shared__ float B[TBH * SA];   // hmin      -> later hmax

  const int tid = threadIdx.x;
  const int gx0 = blockIdx.x * TW;
  const int gy0 = blockIdx.y * TH;
  const int b   = blockIdx.z;
  const float* mslice = mask + (size_t)b * HDIM * WDIM;

  // ---- Pass 0: async-stage raw tile via 16B chunks into LDS; OOB -> +inf ----
  // Staged col c corresponds to global col gx0 + c - 4 (16B-aligned window).
  for (int i = tid; i < TBH * CW; i += NT) {
    int r = i / CW, k = i - r * CW;
    int gy = gy0 + r - 3;
    int gx = gx0 - 4 + 4 * k;            // chunk start; multiple of 4 -> never straddles 0/1024
    unsigned lds_off = (unsigned)(size_t)&A[r * SA + 4 * k];
    if ((unsigned)gy < (unsigned)HDIM && (unsigned)gx < (unsigned)WDIM) {
      const float* gp = mslice + (size_t)gy * WDIM + gx;   // 16B aligned
      // gfx1250 async global->LDS copy (512B/wave-instr), tracked by ASYNCcnt
      asm volatile("global_load_async_to_lds_b128 %0, %1, off"
                   :: "v"(lds_off), "v"(gp) : "memory");
    } else {
      float inf = __builtin_inff();
      *(float4*)&A[r * SA + 4 * k] = make_float4(inf, inf, inf, inf);  // erosion-neutral
    }
  }
  asm volatile("s_wait_asynccnt 0" ::: "memory");
  __syncthreads();

  // ---- Pass 1: horizontal min of 4 thresholded values: B = hmin(thr(A)) ----
  // B[r][c] covers global col gx0 + c - 1 -> reads t cols (c+1)..(c+4).
  for (int i = tid; i < TBH * HC; i += NT) {
    int r = i / HC, c = i - r * HC;
    const float* t = &A[r * SA + c + 1];
    float m = fminf(fminf(thr(t[0]), thr(t[1])), fminf(thr(t[2]), thr(t[3])));
    B[r * SA + c] = m;
  }
  __syncthreads();

  // ---- Pass 2: vertical min of 4: A = eroded; OOB positions -> -inf ----
  // A[r][c] = eroded at global (gy0 + r - 1, gx0 + c - 1).
  for (int i = tid; i < EH * HC; i += NT) {
    int r = i / HC, c = i - r * HC;
    const float* h = &B[r * SA + c];
    float m = fminf(fminf(h[0], h[SA]), fminf(h[2 * SA], h[3 * SA]));
    int gy = gy0 + r - 1, gx = gx0 + c - 1;
    if ((unsigned)gy >= (unsigned)HDIM || (unsigned)gx >= (unsigned)WDIM)
      m = -__builtin_inff();              // dilation-neutral
    A[r * SA + c] = m;
  }
  __syncthreads();

  // ---- Pass 3: horizontal max of 4: B = hmax(A), stride TW ----
  // B[r][x] = max over eroded cols x..x+3 (covers output col gx0 + x).
  for (int i = tid; i < EH * TW; i += NT) {
    int r = i >> 7, c = i & (TW - 1);
    const float* e = &A[r * SA + c];
    float m = fmaxf(fmaxf(e[0], e[1]), fmaxf(e[2], e[3]));
    B[r * TW + c] = m;
  }
  __syncthreads();

  // ---- Pass 4: vertical max of 4, 4 cols/thread -> b128 global store ----
  for (int i = tid; i < (TH * TW) / 4; i += NT) {
    int r = i >> 5, c = (i & 31) * 4;
    const float4 h0 = *(const float4*)&B[(r + 0) * TW + c];
    const float4 h1 = *(const float4*)&B[(r + 1) * TW + c];
    const float4 h2 = *(const float4*)&B[(r + 2) * TW + c];
    const float4 h3 = *(const float4*)&B[(r + 3) * TW + c];
    float4 m;
    m.x = fmaxf(fmaxf(h0.x, h1.x), fmaxf(h2.x, h3.x));
    m.y = fmaxf(fmaxf(h0.y, h1.y), fmaxf(h2.y, h3.y));
    m.z = fmaxf(fmaxf(h0.z, h1.z), fmaxf(h2.z, h3.z));
    m.w = fmaxf(fmaxf(h0.w, h1.w), fmaxf(h2.w, h3.w));
    *(float4*)&out[((size_t)b * HDIM + (gy0 + r)) * WDIM + (gx0 + c)] = m;
  }
}

extern "C" void kernel_launch(void* const* d_in, const int* in_sizes, int n_in,
                              void* d_out, int out_size, void* d_ws, size_t ws_size,
                              hipStream_t stream) {
  const float* score = (const float*)d_in[0];   // (32, 1000) f32
  const float* mask  = (const float*)d_in[1];   // (32, 1024, 1024) f32
  float* out = (float*)d_out;                   // [score | mask] concatenated

  const int sn = in_sizes[0];                   // 32000
  score_thresh_kernel<<<(sn + NT - 1) / NT, NT, 0, stream>>>(score, out, sn);

  dim3 grid(WDIM / TW, HDIM / TH, NBATCH);      // (8, 32, 32)
  open4_kernel<<<grid, NT, 0, stream>>>(mask, out + sn);
}